// GCN_62345745269501
// MI455X (gfx1250) — compile-verified
//
#include <hip/hip_runtime.h>

typedef __attribute__((ext_vector_type(2))) float v2f;
typedef __attribute__((ext_vector_type(8))) float v8f;

#define FDIM 128

// -----------------------------------------------------------------------------
// Kernel 1: support = x @ W   (N x 128) @ (128 x 128), FP32 via WMMA 16x16x4
// One wave -> one 16x16 C tile. Block = 256 threads = 8 waves = 8 column tiles.
// Grid.x = N/16 row tiles.
// -----------------------------------------------------------------------------
__global__ __launch_bounds__(256) void gcn_gemm_wmma(const float* __restrict__ x,
                                                     const float* __restrict__ W,
                                                     float* __restrict__ support,
                                                     int nRows) {
    const int lane    = threadIdx.x & 31;
    const int wave    = threadIdx.x >> 5;   // 0..7 -> column tile
    const int m0      = blockIdx.x * 16;    // row-tile base
    const int n0      = wave * 16;          // col-tile base
    if (m0 >= nRows) return;                // block-uniform: EXEC stays all-ones

    const int halfSel = lane >> 4;          // 0: lanes 0-15, 1: lanes 16-31
    const int l15     = lane & 15;

    const int arow = m0 + l15;              // A-matrix row held by this lane
    const int bcol = n0 + l15;              // B-matrix col held by this lane

    v8f acc = {};
#pragma unroll
    for (int kt = 0; kt < FDIM; kt += 4) {
        const int kb = kt + 2 * halfSel;    // this lane's K pair base
        // A 16x4 f32: VGPR0/1 = K=kb, kb+1 (contiguous -> single b64 load)
        v2f a;
        a.x = x[arow * FDIM + kb];
        a.y = x[arow * FDIM + kb + 1];
        // B 4x16 f32: VGPR0/1 = rows K=kb, kb+1, col = bcol
        v2f b;
        b.x = W[(kb    ) * FDIM + bcol];
        b.y = W[(kb + 1) * FDIM + bcol];
        // 8 args: (neg_a, A, neg_b, B, c_mod, C, reuse_a, reuse_b)
        acc = __builtin_amdgcn_wmma_f32_16x16x4_f32(false, a, false, b,
                                                    (short)0, acc, false, false);
    }

    // C/D layout: VGPR r -> row m0 + r + 8*halfSel, col n0 + l15
#pragma unroll
    for (int r = 0; r < 8; ++r) {
        const int row = m0 + r + 8 * halfSel;
        support[row * FDIM + (n0 + l15)] = acc[r];
    }
}

// -----------------------------------------------------------------------------
// Kernel 2: out = 0.95 * x   (initialize the residual blend before scatter-add)
// -----------------------------------------------------------------------------
__global__ __launch_bounds__(256) void gcn_blend(const float4* __restrict__ x4,
                                                 float4* __restrict__ out4,
                                                 int n4) {
    const int i = blockIdx.x * blockDim.x + threadIdx.x;
    if (i < n4) {
        float4 v = x4[i];
        float4 o;
        o.x = 0.95f * v.x;
        o.y = 0.95f * v.y;
        o.z = 0.95f * v.z;
        o.w = 0.95f * v.w;
        out4[i] = o;
    }
}

// -----------------------------------------------------------------------------
// Kernel 3: for each edge e: out[dst[e]] += 0.05 * w[e] * support[src[e]]
// One wave per edge: 4 floats per lane (float4 gather + 4 f32 atomic adds).
// support & out live in the 192 MB L2 -> gather/scatter is L2-resident.
// -----------------------------------------------------------------------------
__global__ __launch_bounds__(256) void gcn_scatter(const float* __restrict__ support,
                                                   const long long* __restrict__ eidx,
                                                   const float* __restrict__ ew,
                                                   float* __restrict__ out,
                                                   int E) {
    const int wid  = (int)((blockIdx.x * blockDim.x + threadIdx.x) >> 5);
    const int lane = threadIdx.x & 31;
    if (wid >= E) return;   // wave-uniform

    const long long dst = eidx[wid];        // edge_index[0][e]
    const long long src = eidx[(long long)E + wid]; // edge_index[1][e]
    const float     w   = 0.05f * ew[wid];

    const float4* srow = (const float4*)(support + src * FDIM);
    float4 v = srow[lane];

    float* orow = out + dst * FDIM + lane * 4;
    atomicAdd(orow + 0, w * v.x);
    atomicAdd(orow + 1, w * v.y);
    atomicAdd(orow + 2, w * v.z);
    atomicAdd(orow + 3, w * v.w);
}

// -----------------------------------------------------------------------------
extern "C" void kernel_launch(void* const* d_in, const int* in_sizes, int n_in,
                              void* d_out, int out_size, void* d_ws, size_t ws_size,
                              hipStream_t stream) {
    const float*     x    = (const float*)d_in[0];       // [N,128] f32
    const long long* eidx = (const long long*)d_in[1];   // [2,E] i64
    const float*     ew   = (const float*)d_in[2];       // [E]    f32
    const float*     W    = (const float*)d_in[3];       // [128,128] f32
    float*           out  = (float*)d_out;               // [N,128] f32
    float*           support = (float*)d_ws;             // [N,128] f32 scratch

    const int N = in_sizes[0] / FDIM;   // 50000
    const int E = in_sizes[2];          // 800000

    // 1) support = x @ W   (WMMA f32)
    gcn_gemm_wmma<<<(N + 15) / 16, 256, 0, stream>>>(x, W, support, N);

    // 2) out = 0.95 * x
    const int n4 = (N * FDIM) / 4;
    gcn_blend<<<(n4 + 255) / 256, 256, 0, stream>>>((const float4*)x,
                                                    (float4*)out, n4);

    // 3) scatter-add messages: 8 edges per 256-thread block (1 edge per wave)
    const int blocks = (E + 7) / 8;
    gcn_scatter<<<blocks, 256, 0, stream>>>(support, eidx, ew, out, E);
}